// MultiHeadAttention_43550968382074
// MI455X (gfx1250) — compile-verified
//
#include <hip/hip_runtime.h>

// MHA forward for B=4, T=2048, D=1024, H=16, HD=64 (fp32 in/out).
// Matmuls via V_WMMA_F32_16X16X4_F32 (exact fp32 accumulate).
// Global->LDS staging via GLOBAL_LOAD_ASYNC_TO_LDS_B128 with LDS double
// buffering (ASYNCcnt-tracked, overlaps HBM traffic with WMMA compute).

typedef __attribute__((ext_vector_type(2))) float v2f;
typedef __attribute__((ext_vector_type(8))) float v8f;

static __device__ __forceinline__ v8f wmma4(v2f a, v2f b, v8f c) {
  // D = A(16x4) * B(4x16) + C(16x16), fp32
  return __builtin_amdgcn_wmma_f32_16x16x4_f32(
      false, a, false, b, (short)0, c, false, false);
}

// One lane: move 16 bytes global -> LDS, tracked by ASYNCcnt.
static __device__ __forceinline__ void async_ld_b128(unsigned lds_off,
                                                     const float* g) {
  asm volatile("global_load_async_to_lds_b128 %0, %1, off"
               :: "v"(lds_off), "v"(g) : "memory");
}
static __device__ __forceinline__ unsigned lds_off_of(const void* p) {
  return (unsigned)(unsigned long long)p;  // LDS generic addr: offset in [31:0]
}
#define WAIT_ASYNC_0()  asm volatile("s_wait_asynccnt 0" ::: "memory")
#define WAIT_ASYNC_8()  asm volatile("s_wait_asynccnt 8" ::: "memory")
#define WAIT_ASYNC_12() asm volatile("s_wait_asynccnt 12" ::: "memory")

// ---------------------------------------------------------------------------
// C[M,N] = A[M,K] * W[N,K]^T   (A, W row-major; K contiguous for both)
// WG: 256 threads = 8 waves; WG tile 128x256; wave tile 64x64 (4x4 WMMA grid)
// LDS double-buffered, filled by async-to-LDS loads.
// ---------------------------------------------------------------------------
__global__ __launch_bounds__(256) void gemm_nt_wmma(
    const float* __restrict__ A, const float* __restrict__ W,
    float* __restrict__ C, int M, int N, int K) {
  constexpr int KB = 32;
  __shared__ __align__(16) float As[2][128][36];  // stride 36: bank-clean
  __shared__ __align__(16) float Ws[2][256][36];

  const int tid  = threadIdx.x;
  const int lane = tid & 31;
  const int wv   = tid >> 5;     // 0..7
  const int wr   = wv >> 2;      // wave row 0..1  -> 64 rows each
  const int wc   = wv & 3;       // wave col 0..3  -> 64 cols each
  const int half = lane >> 4;    // 0/1
  const int l16  = lane & 15;

  const int m0 = blockIdx.y * 128;
  const int n0 = blockIdx.x * 256;

  v8f acc[4][4];
#pragma unroll
  for (int mt = 0; mt < 4; ++mt)
#pragma unroll
    for (int nt = 0; nt < 4; ++nt) acc[mt][nt] = {};

  const int NP = K / KB;  // panels

  // Async staging: per thread 4 x b128 for A panel, 8 x b128 for W panel.
  auto issue_panel = [&](int p, int k0) {
#pragma unroll
    for (int i = 0; i < 4; ++i) {          // As: 128x32 floats = 1024 chunks
      const int idx = tid + i * 256;
      const int row = idx >> 3, col = (idx & 7) * 4;
      async_ld_b128(lds_off_of(&As[p][row][col]),
                    A + (size_t)(m0 + row) * K + k0 + col);
    }
#pragma unroll
    for (int i = 0; i < 8; ++i) {          // Ws: 256x32 floats = 2048 chunks
      const int idx = tid + i * 256;
      const int row = idx >> 3, col = (idx & 7) * 4;
      async_ld_b128(lds_off_of(&Ws[p][row][col]),
                    W + (size_t)(n0 + row) * K + k0 + col);
    }
  };

  issue_panel(0, 0);
  for (int i = 0; i < NP; ++i) {
    const int p = i & 1;
    const bool nxt = (i + 1) < NP;
    if (nxt) issue_panel(p ^ 1, (i + 1) * KB);  // overlap with compute below
    if (nxt) WAIT_ASYNC_12(); else WAIT_ASYNC_0();  // panel i complete (own)
    __syncthreads();                                // panel i visible to WG

#pragma unroll
    for (int kk = 0; kk < KB; kk += 4) {
      v2f afr[4], bfr[4];
#pragma unroll
      for (int mt = 0; mt < 4; ++mt)  // A frag: lane=row, k = half*2+{0,1}
        afr[mt] = *(const v2f*)&As[p][wr * 64 + mt * 16 + l16][kk + half * 2];
#pragma unroll
      for (int nt = 0; nt < 4; ++nt)  // B frag: lane=col, K halves split
        bfr[nt] = *(const v2f*)&Ws[p][wc * 64 + nt * 16 + l16][kk + half * 2];
#pragma unroll
      for (int mt = 0; mt < 4; ++mt)
#pragma unroll
        for (int nt = 0; nt < 4; ++nt)
          acc[mt][nt] = wmma4(afr[mt], bfr[nt], acc[mt][nt]);
    }
    __syncthreads();  // all waves done reading buf p before it is refilled
  }

  // C/D layout: row = half*8+v, col = l16
#pragma unroll
  for (int mt = 0; mt < 4; ++mt)
#pragma unroll
    for (int nt = 0; nt < 4; ++nt) {
      const int n = n0 + wc * 64 + nt * 16 + l16;
#pragma unroll
      for (int v = 0; v < 8; ++v) {
        const int m = m0 + wr * 64 + mt * 16 + half * 8 + v;
        C[(size_t)m * N + n] = acc[mt][nt][v];
      }
    }
}

// ---------------------------------------------------------------------------
// Flash attention (causal). qkv: [B,T,3D] (q|k|v), y: [B,T,D].
// WG = 128 q rows; 8 waves x 16 q rows. 64-key K/V tiles double-buffered in
// LDS, filled by async-to-LDS loads.
// ---------------------------------------------------------------------------
__global__ __launch_bounds__(256) void attn_wmma(const float* __restrict__ qkv,
                                                 float* __restrict__ y) {
  constexpr int T = 2048, D = 1024, HD = 64;
  constexpr int QB = 128, KT = 64;
  constexpr float NEG_BIG = -3.0e38f;
  const float scale = 0.125f;  // 1/sqrt(64)

  __shared__ __align__(16) float Ks[2][64][68];   // stride 68: bank-clean
  __shared__ __align__(16) float Vs[2][64][68];
  __shared__ __align__(16) float Ps[8][16][68];   // per-wave P staging

  const int tid  = threadIdx.x;
  const int lane = tid & 31;
  const int wv   = tid >> 5;
  const int half = lane >> 4;
  const int l16  = lane & 15;

  const int q0 = blockIdx.x * QB;
  const int h  = blockIdx.y;
  const int b  = blockIdx.z;
  const int qw = q0 + wv * 16;  // this wave's q-row base

  const size_t rs = 3 * (size_t)D;  // 3072
  const float* base = qkv + (size_t)b * T * rs + (size_t)h * HD;

  // preload Q fragments for 16 rows x 64 dims: 16 frags of 16x4
  v2f qf[16];
  {
    const float* qp = base + (size_t)(qw + l16) * rs + half * 2;
#pragma unroll
    for (int kk = 0; kk < 16; ++kk) qf[kk] = *(const v2f*)(qp + kk * 4);
  }

  v8f Oacc[4];
#pragma unroll
  for (int dt = 0; dt < 4; ++dt) Oacc[dt] = {};
  float mrow[8], lsum[8];
#pragma unroll
  for (int v = 0; v < 8; ++v) { mrow[v] = NEG_BIG; lsum[v] = 0.0f; }

  // Async staging of one 64x64 K tile + one V tile: 4+4 b128 per thread.
  auto issue_tile = [&](int p, int key0) {
#pragma unroll
    for (int i = 0; i < 4; ++i) {
      const int idx = tid + i * 256;          // 1024 float4 chunks per tile
      const int row = idx >> 4, col = (idx & 15) * 4;
      const float* kg = base + (size_t)D + (size_t)(key0 + row) * rs + col;
      const float* vg = base + 2 * (size_t)D + (size_t)(key0 + row) * rs + col;
      async_ld_b128(lds_off_of(&Ks[p][row][col]), kg);
      async_ld_b128(lds_off_of(&Vs[p][row][col]), vg);
    }
  };

  const int ntiles = (q0 + QB) / KT;  // uniform per block (causal bound)
  issue_tile(0, 0);
  for (int j = 0; j < ntiles; ++j) {
    const int p = j & 1;
    const bool nxt = (j + 1) < ntiles;
    if (nxt) issue_tile(p ^ 1, (j + 1) * KT);   // overlap with compute
    if (nxt) WAIT_ASYNC_8(); else WAIT_ASYNC_0();
    __syncthreads();                             // tile j visible to WG

    const int key0 = j * KT;

    // S = Q * K^T : 4 tiles of 16x16, K-dim = 64
    v8f S[4];
#pragma unroll
    for (int st = 0; st < 4; ++st) {
      v8f s = {};
#pragma unroll
      for (int kk = 0; kk < 16; ++kk) {
        v2f bf = *(const v2f*)&Ks[p][st * 16 + l16][kk * 4 + half * 2];
        s = wmma4(qf[kk], bf, s);
      }
      S[st] = s;
    }

    // causal mask + scale (C/D layout: row = half*8+v, col = l16)
#pragma unroll
    for (int st = 0; st < 4; ++st) {
      const int key = key0 + st * 16 + l16;
#pragma unroll
      for (int v = 0; v < 8; ++v) {
        const int qr = qw + half * 8 + v;
        S[st][v] = (key <= qr) ? S[st][v] * scale : NEG_BIG;
      }
    }

    // online softmax: row reductions across the 16 lanes holding one row
    float alpha[8];
#pragma unroll
    for (int v = 0; v < 8; ++v) {
      float mx = fmaxf(fmaxf(S[0][v], S[1][v]), fmaxf(S[2][v], S[3][v]));
      mx = fmaxf(mx, __shfl_xor(mx, 1));
      mx = fmaxf(mx, __shfl_xor(mx, 2));
      mx = fmaxf(mx, __shfl_xor(mx, 4));
      mx = fmaxf(mx, __shfl_xor(mx, 8));
      const float mnew = fmaxf(mrow[v], mx);
      alpha[v] = __expf(mrow[v] - mnew);
      mrow[v] = mnew;
      float rsum = 0.0f;
#pragma unroll
      for (int st = 0; st < 4; ++st) {
        const float pexp = __expf(S[st][v] - mnew);
        S[st][v] = pexp;
        rsum += pexp;
      }
      rsum += __shfl_xor(rsum, 1);
      rsum += __shfl_xor(rsum, 2);
      rsum += __shfl_xor(rsum, 4);
      rsum += __shfl_xor(rsum, 8);
      lsum[v] = lsum[v] * alpha[v] + rsum;
    }

    // rescale accumulator
#pragma unroll
    for (int dt = 0; dt < 4; ++dt)
#pragma unroll
      for (int v = 0; v < 8; ++v) Oacc[dt][v] *= alpha[v];

    // P: C-layout -> A-layout via per-wave LDS staging
#pragma unroll
    for (int st = 0; st < 4; ++st)
#pragma unroll
      for (int v = 0; v < 8; ++v)
        Ps[wv][half * 8 + v][st * 16 + l16] = S[st][v];
    asm volatile("s_wait_dscnt 0" ::: "memory");  // wave-local LDS RAW

    // O += P * V : K-dim = 64 keys, 4 output tiles of 16 dims
#pragma unroll
    for (int kk = 0; kk < 16; ++kk) {
      v2f pf = *(const v2f*)&Ps[wv][l16][kk * 4 + half * 2];
#pragma unroll
      for (int dt = 0; dt < 4; ++dt) {
        v2f vf;
        vf.x = Vs[p][kk * 4 + half * 2 + 0][dt * 16 + l16];
        vf.y = Vs[p][kk * 4 + half * 2 + 1][dt * 16 + l16];
        Oacc[dt] = wmma4(pf, vf, Oacc[dt]);
      }
    }
    __syncthreads();  // all waves done with tile j before buffer p is refilled
  }

  // epilogue: O /= l, write y[b, t, h*HD + d]
  float* yb = y + (size_t)b * T * D + (size_t)h * HD;
#pragma unroll
  for (int v = 0; v < 8; ++v) lsum[v] = 1.0f / lsum[v];
#pragma unroll
  for (int dt = 0; dt < 4; ++dt)
#pragma unroll
    for (int v = 0; v < 8; ++v)
      yb[(size_t)(qw + half * 8 + v) * D + dt * 16 + l16] =
          Oacc[dt][v] * lsum[v];
}

// ---------------------------------------------------------------------------
extern "C" void kernel_launch(void* const* d_in, const int* in_sizes, int n_in,
                              void* d_out, int out_size, void* d_ws,
                              size_t ws_size, hipStream_t stream) {
  (void)in_sizes; (void)n_in; (void)out_size; (void)ws_size;
  constexpr int B = 4, T = 2048, D = 1024;
  const float* x    = (const float*)d_in[0];  // [B,T,D]
  const float* Wqkv = (const float*)d_in[1];  // [3D,D]
  const float* Wout = (const float*)d_in[2];  // [D,D]
  float* out = (float*)d_out;                 // [B,T,D]

  float* qkv = (float*)d_ws;                  // [B,T,3D] = 100.7 MB
  float* yat = qkv + (size_t)B * T * 3 * D;   // [B,T,D]  =  33.6 MB

  const dim3 blk(256);
  // 1) qkv = x @ Wqkv^T : M=8192, N=3072, K=1024
  gemm_nt_wmma<<<dim3(3072 / 256, (B * T) / 128), blk, 0, stream>>>(
      x, Wqkv, qkv, B * T, 3 * D, D);
  // 2) causal flash attention
  attn_wmma<<<dim3(T / 128, 16, B), blk, 0, stream>>>(qkv, yat);
  // 3) out = y @ Wout^T : M=8192, N=1024, K=1024
  gemm_nt_wmma<<<dim3(1024 / 256, (B * T) / 128), blk, 0, stream>>>(
      yat, Wout, out, B * T, D, D);
}